// GRC_68599217652438
// MI455X (gfx1250) — compile-verified
//
#include <hip/hip_runtime.h>

// ---------------------------------------------------------------------------
// GRC forward, restructured:
//   y_nat_t = C A^t x0                 (phase A, sequential, 1 WGP, LDS state)
//   u_t     = sum_i M[i] y_nat_{t-i}   (phase B, parallel over t)
//   x_{t+1} = A x_t + B u_t ; cost_t = y'Qy + u'Ru   (phase C, sequential)
// All matmuls via V_WMMA_F32_16X16X4_F32 (native f32 WMMA on CDNA5).
// Batch BT=16 is the WMMA N dimension. Two independent accumulators per tile
// break the WMMA RAW dependency chain (latency-bound sequential phases).
// ---------------------------------------------------------------------------

typedef __attribute__((ext_vector_type(2))) float v2f;
typedef __attribute__((ext_vector_type(8))) float v8f;

#define T_STEPS 512
#define NDIM 128
#define MCDIM 64
#define PDIM 64
#define HDIM 16
#define BTDIM 16

__device__ __forceinline__ v8f wmma_f32(v2f a, v2f b, v8f c) {
  return __builtin_amdgcn_wmma_f32_16x16x4_f32(false, a, false, b, (short)0, c,
                                               false, false);
}

// A-operand fragment (16x4 tile of row-major W, leading dim ld), tile at
// (row0, k0). Layout (ISA 7.12.2): lanes 0-15 -> M=lane, K=k0/k0+1 in v[0]/v[1];
// lanes 16-31 -> M=lane-16, K=k0+2/k0+3.
__device__ __forceinline__ v2f load_a_frag(const float* W, int ld, int row0,
                                           int k0, int lane) {
  int m = lane & 15;
  int ks = k0 + ((lane >> 4) << 1);
  const float* p = W + (row0 + m) * ld + ks;
  v2f a;
  a.x = p[0];
  a.y = p[1];
  return a;
}

// B-operand fragment (4x16 tile) from V stored as V[k*16 + n] (n = trial).
// Mirrored layout: lanes 0-15 -> N=lane, K=k0/k0+1; lanes 16-31 -> K=k0+2/k0+3.
__device__ __forceinline__ v2f load_b_frag16(const float* V, int k0, int lane) {
  int n = lane & 15;
  int ks = k0 + ((lane >> 4) << 1);
  v2f b;
  b.x = V[ks * 16 + n];
  b.y = V[(ks + 1) * 16 + n];
  return b;
}

// C/D 16x16 f32 tile store: VGPR j -> M=j (lanes 0-15), M=j+8 (lanes 16-31).
__device__ __forceinline__ void store_tile16(float* dst, int ldd, int row0,
                                             v8f acc, int lane) {
  int n = lane & 15;
  int mo = (lane >> 4) << 3;
#pragma unroll
  for (int j = 0; j < 8; ++j) dst[(row0 + mo + j) * ldd + n] = acc[j];
}

// 16x16 output tile of W(row-major, ld=K=128) @ V(128x16, V[k*16+n]),
// dual-accumulator over the K=128 reduction.
__device__ __forceinline__ v8f mm_tile_k128(const float* W, const float* V,
                                            int row0, int lane) {
  v8f acc0 = {}, acc1 = {};
#pragma unroll
  for (int k0 = 0; k0 < 64; k0 += 4) {
    v2f a0 = load_a_frag(W, 128, row0, k0, lane);
    v2f b0 = load_b_frag16(V, k0, lane);
    acc0 = wmma_f32(a0, b0, acc0);
    v2f a1 = load_a_frag(W, 128, row0, k0 + 64, lane);
    v2f b1 = load_b_frag16(V, k0 + 64, lane);
    acc1 = wmma_f32(a1, b1, acc1);
  }
  return acc0 + acc1;
}

// ---------------- Phase A: z_{t+1} = A z_t, ynat[t] = C z_t ----------------
__global__ __launch_bounds__(256) void grc_phaseA(const float* __restrict__ A,
                                                  const float* __restrict__ C,
                                                  const float* __restrict__ x0,
                                                  float* __restrict__ ynat) {
  __shared__ float zbuf[2][NDIM * BTDIM];
  const int tid = threadIdx.x;
  const int lane = tid & 31;
  const int wave = tid >> 5;

  for (int idx = tid; idx < NDIM * BTDIM; idx += 256) {
    int n = idx >> 4, b = idx & 15;
    zbuf[0][idx] = x0[b * NDIM + n];  // z[n][b]
  }
  __syncthreads();

  for (int t = 0; t < T_STEPS; ++t) {
    const float* cur = zbuf[t & 1];
    float* nxt = zbuf[(t & 1) ^ 1];

    // z' tile: rows 16*wave .. 16*wave+15, K = 128
    {
      v8f acc = mm_tile_k128(A, cur, wave * 16, lane);
      store_tile16(nxt, 16, wave * 16, acc, lane);
    }
    // y_nat tile (waves 0-3): 64 rows, K = 128; straight to global ws
    if (wave < 4) {
      v8f acc = mm_tile_k128(C, cur, wave * 16, lane);
      store_tile16(ynat + (size_t)t * PDIM * BTDIM, 16, wave * 16, acc, lane);
    }
    __syncthreads();
  }
}

// ---------------- Phase B: u_t = sum_{i<=min(t,H)} M[i] ynat[t-i] ----------
__global__ __launch_bounds__(256) void grc_phaseB(const float* __restrict__ M,
                                                  const float* __restrict__ ynat,
                                                  float* __restrict__ u) {
  const int lane = threadIdx.x & 31;
  const int wave = threadIdx.x >> 5;
  const int t0 = blockIdx.x * 8;           // 64 blocks x 8 timesteps
  const int row0 = (wave & 3) * 16;        // output row tile

  for (int pair = 0; pair < 4; ++pair) {
    const int t = t0 + pair * 2 + (wave >> 2);
    v8f acc0 = {}, acc1 = {};
    const int imax = (t < HDIM) ? t : HDIM;
    for (int i = 0; i <= imax; ++i) {
      const float* Mi = M + (size_t)i * MCDIM * PDIM;
      const float* Yt = ynat + (size_t)(t - i) * PDIM * BTDIM;
#pragma unroll
      for (int k0 = 0; k0 < 32; k0 += 4) {
        v2f a0 = load_a_frag(Mi, PDIM, row0, k0, lane);
        v2f b0 = load_b_frag16(Yt, k0, lane);
        acc0 = wmma_f32(a0, b0, acc0);
        v2f a1 = load_a_frag(Mi, PDIM, row0, k0 + 32, lane);
        v2f b1 = load_b_frag16(Yt, k0 + 32, lane);
        acc1 = wmma_f32(a1, b1, acc1);
      }
    }
    store_tile16(u + (size_t)t * MCDIM * BTDIM, 16, row0, acc0 + acc1, lane);
  }
}

// ---------------- Phase C: x recurrence + per-step cost --------------------
__global__ __launch_bounds__(256) void grc_phaseC(
    const float* __restrict__ A, const float* __restrict__ B,
    const float* __restrict__ C, const float* __restrict__ Q,
    const float* __restrict__ R, const float* __restrict__ x0,
    const float* __restrict__ u, float* __restrict__ out) {
  __shared__ float xbuf[2][NDIM * BTDIM];
  __shared__ float ybuf[PDIM * BTDIM];
  __shared__ float ubuf[MCDIM * BTDIM];
  __shared__ float cpart[BTDIM];
  const int tid = threadIdx.x;
  const int lane = tid & 31;
  const int wave = tid >> 5;

  for (int idx = tid; idx < NDIM * BTDIM; idx += 256) {
    int n = idx >> 4, b = idx & 15;
    xbuf[0][idx] = x0[b * NDIM + n];
  }
  __syncthreads();

  for (int t = 0; t < T_STEPS; ++t) {
    const float* cur = xbuf[t & 1];
    float* nxt = xbuf[(t & 1) ^ 1];
    const float* ut = u + (size_t)t * MCDIM * BTDIM;

    for (int idx = tid; idx < MCDIM * BTDIM; idx += 256) ubuf[idx] = ut[idx];
    if (tid < BTDIM) cpart[tid] = 0.0f;

    // x' = A@x + B@u_t  (8 tiles, one per wave), dual accumulators
    {
      const int row0 = wave * 16;
      v8f acc0 = {}, acc1 = {};
#pragma unroll
      for (int k0 = 0; k0 < 64; k0 += 4) {
        v2f a0 = load_a_frag(A, NDIM, row0, k0, lane);
        v2f b0 = load_b_frag16(cur, k0, lane);
        acc0 = wmma_f32(a0, b0, acc0);
        v2f a1 = load_a_frag(A, NDIM, row0, k0 + 64, lane);
        v2f b1 = load_b_frag16(cur, k0 + 64, lane);
        acc1 = wmma_f32(a1, b1, acc1);
      }
#pragma unroll
      for (int k0 = 0; k0 < 64; k0 += 8) {
        v2f a0 = load_a_frag(B, MCDIM, row0, k0, lane);
        v2f b0 = load_b_frag16(ut, k0, lane);  // from L2, no sync needed
        acc0 = wmma_f32(a0, b0, acc0);
        v2f a1 = load_a_frag(B, MCDIM, row0, k0 + 4, lane);
        v2f b1 = load_b_frag16(ut, k0 + 4, lane);
        acc1 = wmma_f32(a1, b1, acc1);
      }
      store_tile16(nxt, 16, row0, acc0 + acc1, lane);
    }
    // y_obs = C@x (waves 0-3)
    if (wave < 4) {
      v8f acc = mm_tile_k128(C, cur, wave * 16, lane);
      store_tile16(ybuf, 16, wave * 16, acc, lane);
    }
    __syncthreads();

    // QY (waves 0-3) / RU (waves 4-7), then dot columns against y / u
    {
      const int row0 = (wave & 3) * 16;
      const float* W = (wave < 4) ? Q : R;
      const float* V = (wave < 4) ? ybuf : ubuf;
      v8f acc0 = {}, acc1 = {};
#pragma unroll
      for (int k0 = 0; k0 < 32; k0 += 4) {
        v2f a0 = load_a_frag(W, PDIM, row0, k0, lane);
        v2f b0 = load_b_frag16(V, k0, lane);
        acc0 = wmma_f32(a0, b0, acc0);
        v2f a1 = load_a_frag(W, PDIM, row0, k0 + 32, lane);
        v2f b1 = load_b_frag16(V, k0 + 32, lane);
        acc1 = wmma_f32(a1, b1, acc1);
      }
      v8f acc = acc0 + acc1;
      const int n = lane & 15;
      const int mo = (lane >> 4) << 3;
      float p = 0.0f;
#pragma unroll
      for (int j = 0; j < 8; ++j) p += acc[j] * V[(row0 + mo + j) * 16 + n];
      atomicAdd(&cpart[n], p);
    }
    __syncthreads();

    if (tid == 0) {
      float s = 0.0f;
      for (int b = 0; b < BTDIM; ++b) s += cpart[b];
      out[t] = s * (1.0f / (float)BTDIM);
    }
    __syncthreads();  // protects cpart/ubuf/ybuf rewrite + x buffer swap
  }
}

extern "C" void kernel_launch(void* const* d_in, const int* in_sizes, int n_in,
                              void* d_out, int out_size, void* d_ws,
                              size_t ws_size, hipStream_t stream) {
  const float* A = (const float*)d_in[0];   // (128,128)
  const float* B = (const float*)d_in[1];   // (128,64)
  const float* C = (const float*)d_in[2];   // (64,128)
  const float* Q = (const float*)d_in[3];   // (64,64)
  const float* R = (const float*)d_in[4];   // (64,64)
  const float* M = (const float*)d_in[5];   // (17,64,64)
  const float* x0 = (const float*)d_in[6];  // (16,128,1)

  float* ynat = (float*)d_ws;                                 // 512*64*16 f32
  float* u = ynat + (size_t)T_STEPS * PDIM * BTDIM;           // 512*64*16 f32

  grc_phaseA<<<1, 256, 0, stream>>>(A, C, x0, ynat);
  grc_phaseB<<<64, 256, 0, stream>>>(M, ynat, u);
  grc_phaseC<<<1, 256, 0, stream>>>(A, B, C, Q, R, x0, u, (float*)d_out);
}